// CRF_16071767622429
// MI455X (gfx1250) — compile-verified
//
#include <hip/hip_runtime.h>
#include <hip/hip_bf16.h>

typedef __attribute__((ext_vector_type(2))) float v2f;
typedef __attribute__((ext_vector_type(8))) float v8f;

#define T_TAGS 21
#define T_PAD  32
#define B_BATCH 64
#define S_LEN  512
#define H_DIM  1024
#define KTILES (H_DIM / 4)   // 256 K-steps of 4

// ---------------------------------------------------------------------------
// Pack W (1024 x 21, row-major) into WMMA B-fragment layout for f32 16x16x4:
// per (kt, ntile): lane L holds N = ntile*16 + L%16, K = kt*4 + (L/16)*2 and K+1.
// Bpack[(kt*2+ntile)*32 + lane] = { W[K][N], W[K+1][N] }   (zeros for N >= 21)
// ---------------------------------------------------------------------------
__global__ void crf_packW(const float* __restrict__ W, float* __restrict__ Bpack) {
    int idx = blockIdx.x * blockDim.x + threadIdx.x;   // 0 .. KTILES*2*32-1
    if (idx >= KTILES * 2 * 32) return;
    int lane  = idx & 31;
    int ntile = (idx >> 5) & 1;
    int kt    = idx >> 6;
    int n = ntile * 16 + (lane & 15);
    int k = kt * 4 + ((lane >> 4) << 1);
    float v0 = 0.f, v1 = 0.f;
    if (n < T_TAGS) {
        v0 = W[(size_t)k * T_TAGS + n];
        v1 = W[(size_t)(k + 1) * T_TAGS + n];
    }
    Bpack[(size_t)idx * 2 + 0] = v0;
    Bpack[(size_t)idx * 2 + 1] = v1;
}

// ---------------------------------------------------------------------------
// logits[m][0..31] = x[m][:] @ W + b   (cols 21..31 junk, never read downstream)
// One wave per 16-row tile; f32 WMMA 16x16x4, K-loop of 256.
// ---------------------------------------------------------------------------
__global__ __launch_bounds__(256) void crf_gemm(
    const float* __restrict__ x, const float* __restrict__ Bpack,
    const float* __restrict__ bias, float* __restrict__ logits) {
    int wave = (blockIdx.x * blockDim.x + threadIdx.x) >> 5;
    int lane = threadIdx.x & 31;
    int m0   = wave * 16;
    int mrow = m0 + (lane & 15);
    int khalf = (lane >> 4) << 1;                       // 0 or 2
    const float* arow = x + (size_t)mrow * H_DIM + khalf;
    const v2f* bp = (const v2f*)Bpack;

    v8f c0 = {}; v8f c1 = {};
#pragma unroll 4
    for (int kt = 0; kt < KTILES; ++kt) {
        v2f a  = *(const v2f*)(arow + kt * 4);          // A frag: {x[m][k], x[m][k+1]}
        v2f b0 = bp[(kt * 2 + 0) * 32 + lane];
        v2f b1 = bp[(kt * 2 + 1) * 32 + lane];
        c0 = __builtin_amdgcn_wmma_f32_16x16x4_f32(false, a, false, b0,
                                                   (short)0, c0, false, false);
        c1 = __builtin_amdgcn_wmma_f32_16x16x4_f32(false, a, false, b1,
                                                   (short)0, c1, false, false);
    }

    int n0 = lane & 15;
    float bias0 = (n0 < T_TAGS) ? bias[n0] : 0.f;
    float bias1 = (n0 + 16 < T_TAGS) ? bias[n0 + 16] : 0.f;
    int mbase = m0 + ((lane >> 4) << 3);                // lanes 16-31 hold rows +8
#pragma unroll
    for (int r = 0; r < 8; ++r) {
        float* row = logits + (size_t)(mbase + r) * T_PAD;
        row[n0]      = c0[r] + bias0;
        row[n0 + 16] = c1[r] + bias1;
    }
}

// ---------------------------------------------------------------------------
// Forward algorithm + numerator: one wave32 per batch, lane j = state j.
// llpb[b] = num - denom
// ---------------------------------------------------------------------------
__global__ __launch_bounds__(32) void crf_forward(
    const float* __restrict__ logits, const int* __restrict__ x_mask,
    const int* __restrict__ tags, const float* __restrict__ trans,
    const float* __restrict__ st, const float* __restrict__ et,
    float* __restrict__ llpb) {
    const float NEG_INF = -1e30f;
    int b = blockIdx.x;
    int j = threadIdx.x;

    float Tcol[T_TAGS];
#pragma unroll
    for (int i = 0; i < T_TAGS; ++i)
        Tcol[i] = (j < T_TAGS) ? trans[i * T_TAGS + j] : 0.f;

    const float* lg = logits + (size_t)b * S_LEN * T_PAD;
    const int* mk = x_mask + b * S_LEN;
    const int* tg = tags + b * S_LEN;

    float alpha = (j < T_TAGS) ? (st[j] + lg[j]) : NEG_INF;

    for (int t = 1; t < S_LEN; ++t) {
        float v[T_TAGS];
        float mx = NEG_INF;
#pragma unroll
        for (int i = 0; i < T_TAGS; ++i) {
            float ai = __shfl(alpha, i, 32);
            v[i] = ai + Tcol[i];
            mx = fmaxf(mx, v[i]);
        }
        float s = 0.f;
#pragma unroll
        for (int i = 0; i < T_TAGS; ++i) s += expf(v[i] - mx);
        float nw = mx + logf(s) + lg[(size_t)t * T_PAD + j];
        if (mk[t] != 0 && j < T_TAGS) alpha = nw;
    }

    // denom = logsumexp_j(alpha + et[j])
    float aj = (j < T_TAGS) ? (alpha + et[j]) : NEG_INF;
    float mx = aj;
    for (int o = 16; o > 0; o >>= 1) mx = fmaxf(mx, __shfl_xor(mx, o, 32));
    float ex = (j < T_TAGS) ? expf(aj - mx) : 0.f;
    for (int o = 16; o > 0; o >>= 1) ex += __shfl_xor(ex, o, 32);
    float denom = mx + logf(ex);

    // numerator partial sums (emit + internal transitions + mask count)
    float part = 0.f;
    int cnt = 0;
    for (int t = j; t < S_LEN; t += 32) {
        int m = mk[t];
        cnt += (m != 0) ? 1 : 0;
        if (m != 0) {
            part += lg[(size_t)t * T_PAD + tg[t]];
            if (t > 0) part += trans[tg[t - 1] * T_TAGS + tg[t]];
        }
    }
    for (int o = 16; o > 0; o >>= 1) {
        part += __shfl_xor(part, o, 32);
        cnt  += __shfl_xor(cnt, o, 32);
    }
    if (j == 0) {
        int last_idx = cnt - 1;
        int last_tag = tg[last_idx];
        float num = st[tg[0]] + part + et[last_tag];
        llpb[b] = num - denom;
    }
}

// ---------------------------------------------------------------------------
// Constrained Viterbi + backtrace: one wave32 per batch, backpointers in LDS.
// Writes predicted tags (as float) to out[b*S .. b*S+S-1].
// ---------------------------------------------------------------------------
__global__ __launch_bounds__(32) void crf_viterbi(
    const float* __restrict__ logits, const int* __restrict__ x_mask,
    const float* __restrict__ trans, const float* __restrict__ st,
    const float* __restrict__ et, const int* __restrict__ at,
    const int* __restrict__ ast, const int* __restrict__ aen,
    float* __restrict__ out) {
    __shared__ int bp[S_LEN - 1][32];                   // 511*32*4 = 65,408 B LDS
    const float NEG_INF = -1e30f;
    const float NEG = -10000.0f;
    int b = blockIdx.x;
    int j = threadIdx.x;

    float Tcol[T_TAGS];
#pragma unroll
    for (int i = 0; i < T_TAGS; ++i)
        Tcol[i] = (j < T_TAGS && at[i * T_TAGS + j] != 0) ? trans[i * T_TAGS + j]
                                                          : NEG;

    const float* lg = logits + (size_t)b * S_LEN * T_PAD;
    const int* mk = x_mask + b * S_LEN;

    float va = (j < T_TAGS) ? (((ast[j] != 0) ? st[j] : NEG) + lg[j]) : NEG_INF;

    for (int t = 1; t < S_LEN; ++t) {
        float best = NEG_INF;
        int arg = 0;
#pragma unroll
        for (int i = 0; i < T_TAGS; ++i) {
            float sc = __shfl(va, i, 32) + Tcol[i];
            if (sc > best) { best = sc; arg = i; }      // first max kept (>)
        }
        float nw = best + lg[(size_t)t * T_PAD + j];
        int keep = (mk[t] != 0);
        int bpv;
        if (keep && j < T_TAGS) { va = nw; bpv = arg; } else { bpv = j; }
        bp[t - 1][j] = bpv;
    }

    // last_tag = first-index argmax_j(va + ev[j])
    float sc = (j < T_TAGS) ? (va + ((aen[j] != 0) ? et[j] : NEG)) : NEG_INF;
    int idx = j;
    for (int o = 16; o > 0; o >>= 1) {
        float sv = __shfl_xor(sc, o, 32);
        int si   = __shfl_xor(idx, o, 32);
        if (sv > sc || (sv == sc && si < idx)) { sc = sv; idx = si; }
    }
    __syncthreads();                                    // flush LDS bp writes

    if (j == 0) {
        int tag = idx;
        float* row = out + (size_t)b * S_LEN;
        row[S_LEN - 1] = (float)tag;
        for (int t = S_LEN - 2; t >= 0; --t) {
            tag = bp[t][tag];
            row[t] = (float)tag;
        }
    }
}

// ---------------------------------------------------------------------------
// Deterministic sum of 64 per-batch log-likelihoods -> out[B*S]
// ---------------------------------------------------------------------------
__global__ __launch_bounds__(32) void crf_reduce(const float* __restrict__ llpb,
                                                 float* __restrict__ out) {
    int j = threadIdx.x;
    float s = llpb[j] + llpb[j + 32];
    for (int o = 16; o > 0; o >>= 1) s += __shfl_xor(s, o, 32);
    if (j == 0) out[0] = s;
}

// ---------------------------------------------------------------------------
extern "C" void kernel_launch(void* const* d_in, const int* in_sizes, int n_in,
                              void* d_out, int out_size, void* d_ws, size_t ws_size,
                              hipStream_t stream) {
    const float* x      = (const float*)d_in[0];   // (64,512,1024)
    const int*   x_mask = (const int*)d_in[1];     // (64,512) bool
    const int*   tags   = (const int*)d_in[2];     // (64,512) i32
    const float* W      = (const float*)d_in[3];   // (1024,21)
    const float* bias   = (const float*)d_in[4];   // (21,)
    const float* trans  = (const float*)d_in[5];   // (21,21)
    const float* st     = (const float*)d_in[6];   // (21,)
    const float* et     = (const float*)d_in[7];   // (21,)
    const int*   at     = (const int*)d_in[8];     // (21,21) bool
    const int*   ast    = (const int*)d_in[9];     // (21,) bool
    const int*   aen    = (const int*)d_in[10];    // (21,) bool
    float* out = (float*)d_out;                    // 64*512 tags + 1 ll

    // workspace layout
    float* logits = (float*)d_ws;                        // 32768*32 f32 = 4 MB
    float* Bpack  = logits + (size_t)B_BATCH * S_LEN * T_PAD;  // 128 KB
    float* llpb   = Bpack + (size_t)KTILES * 2 * 32 * 2;       // 64 f32

    crf_packW<<<dim3((KTILES * 2 * 32 + 255) / 256), dim3(256), 0, stream>>>(W, Bpack);
    // 2048 row-tiles of 16, 8 waves/block -> 256 blocks
    crf_gemm<<<dim3(256), dim3(256), 0, stream>>>(x, Bpack, bias, logits);
    crf_forward<<<dim3(B_BATCH), dim3(32), 0, stream>>>(logits, x_mask, tags, trans,
                                                        st, et, llpb);
    crf_viterbi<<<dim3(B_BATCH), dim3(32), 0, stream>>>(logits, x_mask, trans, st,
                                                        et, at, ast, aen, out);
    crf_reduce<<<dim3(1), dim3(32), 0, stream>>>(llpb, out + (size_t)B_BATCH * S_LEN);
}